// DPS_55336358642719
// MI455X (gfx1250) — compile-verified
//
#include <hip/hip_runtime.h>
#include <hip/hip_bf16.h>

typedef __attribute__((ext_vector_type(2))) float v2f;
typedef __attribute__((ext_vector_type(8))) float v8f;

#define H_SCORE 16
#define W_SCORE 16
#define NSAMP   500
#define TOPK    16
#define SIGMA   0.05f

__device__ __forceinline__ unsigned hashu(unsigned x) {
  x ^= x >> 16; x *= 0x7feb352dU;
  x ^= x >> 15; x *= 0x846ca68bU;
  x ^= x >> 16;
  return x;
}

__device__ __forceinline__ int imax(int a, int b) { return a > b ? a : b; }
__device__ __forceinline__ int imin(int a, int b) { return a < b ? a : b; }

// ---------------------------------------------------------------------------
// Kernel 1: perturbed top-K indicator histogram.
// grid = 8 (batch), block = 256 (8 wave32s, one sample per wave per iter).
// Writes ind[b,k,d] (8*16*256 f32) and a u64 nonzero-chunk bitmask per batch.
// ---------------------------------------------------------------------------
__global__ __launch_bounds__(256) void dps_ind_kernel(
    const float* __restrict__ scores, float* __restrict__ ind,
    unsigned long long* __restrict__ flags64) {
  __shared__ float s_norm[256];
  __shared__ float redA[256];
  __shared__ float redB[256];
  __shared__ int   counts[TOPK * 256];
  __shared__ int   colsum[256];
  __shared__ unsigned fmask[2];

  const int t    = threadIdx.x;
  const int b    = blockIdx.x;
  const int lane = t & 31;
  const int wave = t >> 5;

  // normalize scores (min/max over 256)
  const float s = scores[b * 256 + t];
  redA[t] = s; redB[t] = s;
  if (t < 2) fmask[t] = 0;
  __syncthreads();
  for (int str = 128; str > 0; str >>= 1) {
    if (t < str) {
      redA[t] = fminf(redA[t], redA[t + str]);
      redB[t] = fmaxf(redB[t], redB[t + str]);
    }
    __syncthreads();
  }
  const float smin = redA[0], smax = redB[0];
  s_norm[t] = (s - smin) / (smax - smin + 1e-5f);
  for (int k = 0; k < TOPK; ++k) counts[k * 256 + t] = 0;
  __syncthreads();

  for (int smp = wave; smp < NSAMP; smp += 8) {
    // lane owns contiguous indices [8*lane, 8*lane+8)
    const int base = lane * 8;
    float v[8];
#pragma unroll
    for (int j = 0; j < 8; ++j) {
      const unsigned seed = ((unsigned)b * 0x9E3779B9u) ^
                            ((unsigned)smp * 0x85EBCA6Bu) ^
                            ((unsigned)(base + j) * 0xC2B2AE35u);
      const unsigned h1 = hashu(seed ^ 0x68bc21ebu);
      const unsigned h2 = hashu(seed ^ 0x02e5be93u);
      const float u1 = ((float)h1 + 0.5f) * 2.3283064365e-10f;
      const float u2 = (float)h2 * 2.3283064365e-10f;
      const float g  = sqrtf(-2.0f * __logf(u1)) * __cosf(6.28318530718f * u2);
      v[j] = s_norm[base + j] + SIGMA * g;
    }
    // 16 rounds of wave-wide argmax; zap winners (zap mask == top-16 set)
    unsigned zap = 0;
#pragma unroll 1
    for (int r = 0; r < TOPK; ++r) {
      float bv = -3.0e38f;
      int   bi = -1;
#pragma unroll
      for (int j = 0; j < 8; ++j)
        if (!((zap >> j) & 1u) && v[j] > bv) { bv = v[j]; bi = base + j; }
      for (int off = 16; off > 0; off >>= 1) {
        const float ov = __shfl_xor(bv, off, 32);
        const int   oi = __shfl_xor(bi, off, 32);
        if (ov > bv || (ov == bv && oi < bi)) { bv = ov; bi = oi; }
      }
      if ((bi >> 3) == lane) zap |= 1u << (bi & 7);
    }
    // rank members in ascending-index order (wave prefix sum of popcounts)
    const int cnt = __popc(zap);
    int incl = cnt;
    for (int off = 1; off < 32; off <<= 1) {
      const int n = __shfl_up(incl, off, 32);
      if (lane >= off) incl += n;
    }
    int rank = incl - cnt;
#pragma unroll
    for (int j = 0; j < 8; ++j) {
      if ((zap >> j) & 1u) {
        atomicAdd(&counts[rank * 256 + base + j], 1);
        ++rank;
      }
    }
  }
  __syncthreads();

  int cs = 0;
  for (int k = 0; k < TOPK; ++k) {
    const int cv = counts[k * 256 + t];
    cs += cv;
    ind[(b * TOPK + k) * 256 + t] = (float)cv * (1.0f / (float)NSAMP);
  }
  colsum[t] = cs;
  __syncthreads();
  if (t < 64) {
    const int f = colsum[4 * t] | colsum[4 * t + 1] |
                  colsum[4 * t + 2] | colsum[4 * t + 3];
    if (f != 0) atomicOr(&fmask[t >> 5], 1u << (t & 31));
  }
  __syncthreads();
  if (t == 0)
    flags64[b] = ((unsigned long long)fmask[1] << 32) | (unsigned long long)fmask[0];
}

// ---------------------------------------------------------------------------
// One WMMA K-chunk (4 K-values = 4 consecutive w at one i).
// CLO_X: low-side column clamp needed (.x comp, only w0==0 chunk).
// CHI_Y: high-side column clamp needed (.y comp, only w0==12 chunk).
// Middle chunks compile with zero bounds checks.
// ---------------------------------------------------------------------------
template <bool CLO_X, bool CHI_Y>
__device__ __forceinline__ void wmma_chunk(const float* __restrict__ xrow,
                                           const float* __restrict__ indB,
                                           int iw0, int colb, v8f acc[4]) {
  v2f a;
  a.x = indB[iw0];
  a.y = indB[iw0 + 1];
  v2f bt[4];
#pragma unroll
  for (int tq = 0; tq < 4; ++tq) {
    const int c0 = colb + 16 * tq;       // column for .x (row K = 2*half)
    const int c1 = c0 + 64;              // column for .y (row K = 2*half+1)
    float vx = xrow[CLO_X ? imax(c0, 0) : c0];
    float vy = xrow[CHI_Y ? imin(c1, 1023) : c1];
    if (CLO_X) vx = (c0 >= 0) ? vx : 0.0f;
    if (CHI_Y) vy = (c1 < 1024) ? vy : 0.0f;
    bt[tq].x = vx;
    bt[tq].y = vy;
  }
#pragma unroll
  for (int tq = 0; tq < 4; ++tq)
    acc[tq] = __builtin_amdgcn_wmma_f32_16x16x4_f32(
        false, a, false, bt[tq], (short)0, acc[tq], false, false);
}

// ---------------------------------------------------------------------------
// Kernel 2: out(16k x 16384) = ind(16k x 256) @ gathered_patches(256 x 16384)
// per (b,c) via V_WMMA_F32_16X16X4_F32. One wave = 16(k) x 64(q) strip at
// fixed (b,c,p). Tile coords forced into SGPRs (readfirstlane) so zero-chunk
// skips and row-OOB skips are pure scalar branches; EXEC never changes.
// ---------------------------------------------------------------------------
__global__ __launch_bounds__(256) void dps_wmma_kernel(
    const float* __restrict__ x, const float* __restrict__ ind,
    const unsigned long long* __restrict__ flags64, float* __restrict__ out) {
  // wave-uniform tile id, forced scalar
  const int wave = __builtin_amdgcn_readfirstlane(
      (int)(blockIdx.x * 8 + (threadIdx.x >> 5)));
  const int lane = threadIdx.x & 31;
  const int half = lane >> 4;   // 0: lanes 0-15, 1: lanes 16-31
  const int ln   = lane & 15;

  const int q0 = (wave & 1) * 64;        // column strip base   (scalar)
  const int p  = (wave >> 1) & 127;      // patch row           (scalar)
  const int c  = (wave >> 8) % 3;        // channel             (scalar)
  const int b  = wave / 768;             // batch               (scalar)

  v8f acc[4] = {};

  const float* indB = ind + (b * TOPK + ln) * 256;     // A row for this lane
  const float* xc   = x + (((size_t)(b * 3 + c)) << 20);
  const unsigned long long m = flags64[b];             // scalar bitmask

  const int colb0 = q0 - 32 + 128 * half + ln;         // w0=0 column base

  for (int i = 0; i < H_SCORE; ++i) {
    const unsigned mi = (unsigned)(m >> (i * 4)) & 0xFu;
    if (mi == 0) continue;                             // scalar skip
    const int row = i * 64 + p - 32;                   // unpadded source row
    if ((unsigned)row >= 1024u) continue;              // scalar skip (zero slab)
    const float* xrow = xc + (size_t)row * 1024;
    const int iwb = i * 16 + 2 * half;

    if (mi & 1u) wmma_chunk<true,  false>(xrow, indB, iwb + 0,  colb0 + 0,   acc);
    if (mi & 2u) wmma_chunk<false, false>(xrow, indB, iwb + 4,  colb0 + 256, acc);
    if (mi & 4u) wmma_chunk<false, false>(xrow, indB, iwb + 8,  colb0 + 512, acc);
    if (mi & 8u) wmma_chunk<false, true >(xrow, indB, iwb + 12, colb0 + 768, acc);
  }

  // D layout: VGPR v -> k = v + 8*half, col = ln. out[(b*16+k), c, p, q]
  float* o = out + ((size_t)((b * TOPK + 8 * half) * 3 + c)) * 16384 +
             p * 128 + q0 + ln;
#pragma unroll
  for (int v = 0; v < 8; ++v) {
    const size_t ko = (size_t)v * 3 * 16384;
    o[ko + 0]  = acc[0][v];
    o[ko + 16] = acc[1][v];
    o[ko + 32] = acc[2][v];
    o[ko + 48] = acc[3][v];
  }
}

// ---------------------------------------------------------------------------
extern "C" void kernel_launch(void* const* d_in, const int* in_sizes, int n_in,
                              void* d_out, int out_size, void* d_ws, size_t ws_size,
                              hipStream_t stream) {
  const float* x_high   = (const float*)d_in[0];  // (8,3,1024,1024)
  const float* scores2d = (const float*)d_in[1];  // (8,16,16)
  float* out = (float*)d_out;                     // (128,3,128,128)

  float* ind_ws = (float*)d_ws;                   // 8*16*256 f32 = 131072 B
  unsigned long long* flags_ws =
      (unsigned long long*)((char*)d_ws + 8 * 16 * 256 * sizeof(float)); // 8 u64

  dps_ind_kernel<<<8, 256, 0, stream>>>(scores2d, ind_ws, flags_ws);

  // 8 b * 3 c * 128 p * 2 qstrips = 6144 waves -> 768 blocks of 8 waves
  dps_wmma_kernel<<<768, 256, 0, stream>>>(x_high, ind_ws, flags_ws, out);

  (void)in_sizes; (void)n_in; (void)out_size; (void)ws_size;
}